// JPEGCompression_25881472926199
// MI455X (gfx1250) — compile-verified
//
#include <hip/hip_runtime.h>
#include <cstdint>

// JPEG-compression noise simulation, MI455X (gfx1250, wave32).
//   out = clip(img + pixel_noise + row_boundary_noise + col_boundary_noise, 0, 1)
// Memory-bound floor ~17us (402MB @ 23.3TB/s); RNG compute is overlapped with
// the image loads via the CDNA5 async global->LDS path (ASYNCcnt).

namespace {
constexpr uint32_t Wd = 512u;   // image width
constexpr uint32_t Hd = 512u;   // image height
constexpr int TPB = 256;        // 8 wave32 per block
}

typedef float v4f __attribute__((ext_vector_type(4)));   // native 128-bit vec

__device__ __forceinline__ uint32_t rotl32(uint32_t x, int r) {
  return (x << r) | (x >> (32 - r));   // folds to v_alignbit_b32
}

struct U2 { uint32_t a, b; };

// threefry2x32, 20 rounds (JAX's PRNG primitive). Counter (c0,c1) -> 2x u32.
__device__ __forceinline__ U2 threefry2x32(uint32_t k0, uint32_t k1,
                                           uint32_t c0, uint32_t c1) {
  uint32_t k2 = k0 ^ k1 ^ 0x1BD11BDAu;
  uint32_t x0 = c0 + k0, x1 = c1 + k1;
#define TF_R4(ra, rb, rc, rd)                              \
  x0 += x1; x1 = rotl32(x1, ra); x1 ^= x0;                 \
  x0 += x1; x1 = rotl32(x1, rb); x1 ^= x0;                 \
  x0 += x1; x1 = rotl32(x1, rc); x1 ^= x0;                 \
  x0 += x1; x1 = rotl32(x1, rd); x1 ^= x0;
  TF_R4(13, 15, 26, 6)  x0 += k1; x1 += k2 + 1u;
  TF_R4(17, 29, 16, 24) x0 += k2; x1 += k0 + 2u;
  TF_R4(13, 15, 26, 6)  x0 += k0; x1 += k1 + 3u;
  TF_R4(17, 29, 16, 24) x0 += k1; x1 += k2 + 4u;
  TF_R4(13, 15, 26, 6)  x0 += k2; x1 += k0 + 5u;
#undef TF_R4
  return {x0, x1};
}

// Giles (2010) single-precision erfinv: FMAs + one fast log.
__device__ __forceinline__ float erfinv_f(float x) {
  float w = -__logf(fmaf(-x, x, 1.0f));
  float p;
  if (w < 5.0f) {
    w -= 2.5f;
    p =              2.81022636e-08f;
    p = fmaf(p, w,   3.43273939e-07f);
    p = fmaf(p, w,  -3.52338770e-06f);
    p = fmaf(p, w,  -4.39150654e-06f);
    p = fmaf(p, w,   2.18580870e-04f);
    p = fmaf(p, w,  -1.25372503e-03f);
    p = fmaf(p, w,  -4.17768164e-03f);
    p = fmaf(p, w,   2.46640727e-01f);
    p = fmaf(p, w,   1.50140941e+00f);
  } else {
    w = __fsqrt_rn(w) - 3.0f;
    p =             -2.00214257e-04f;
    p = fmaf(p, w,   1.00950558e-04f);
    p = fmaf(p, w,   1.34934322e-03f);
    p = fmaf(p, w,  -3.67342844e-03f);
    p = fmaf(p, w,   5.73950773e-03f);
    p = fmaf(p, w,  -7.62246130e-03f);
    p = fmaf(p, w,   9.43887047e-03f);
    p = fmaf(p, w,   1.00167406e+00f);
    p = fmaf(p, w,   2.83297682e+00f);
  }
  return p * x;
}

// u32 -> standard normal: uniform in (-1,1), sqrt(2)*erfinv(u)  (JAX scheme).
__device__ __forceinline__ float u32_normal(uint32_t bits) {
  float u01 = ((float)(bits >> 8) + 0.5f) * 0x1p-24f;  // (0,1]: max input rounds to exactly 1.0f
  float u = fmaf(u01, 2.0f, -1.0f);
  return 1.41421356237f * erfinv_f(u);
}

__device__ __forceinline__ float clamp01(float x) {
  return fminf(fmaxf(x, 0.0f), 1.0f);   // v_med3-able
}

// Sub-keys for the three independent noise streams (stand-ins for
// jax.random.split(key(42), 3)); fixed constants -> deterministic output.
#define KP0 0x2f1a6b93u
#define KP1 0x0000002au
#define KR0 0x85ebca6bu
#define KR1 0xc2b2ae35u
#define KC0 0x27d4eb2fu
#define KC1 0x165667b1u

__global__ __launch_bounds__(TPB) void
JPEGCompression_25881472926199_kernel(const float* __restrict__ img,
                                      const int* __restrict__ qptr,
                                      float* __restrict__ out,
                                      uint32_t n4) {
  __shared__ v4f tile[TPB];

  uint32_t idx4 = blockIdx.x * TPB + threadIdx.x;
  if (idx4 >= n4) return;
  uint32_t e0 = idx4 << 2;                 // first of 4 consecutive elements

  // ---- Kick off the async DMA of our quad into LDS (ASYNCcnt++). The RNG
  // below is data-independent, so the whole load latency hides under it.
  uint32_t lds_off = (uint32_t)(uintptr_t)&tile[threadIdx.x];
  const float* gp = img + e0;
  asm volatile("global_load_async_to_lds_b128 %0, %1, off"
               :: "v"(lds_off), "v"(gp) : "memory");

  // ---- Coordinates (all power-of-two except B*C, which we never split).
  uint32_t w  = e0 & (Wd - 1u);            // multiple of 4
  uint32_t h  = (e0 >> 9) & (Hd - 1u);
  uint32_t bc = e0 >> 18;                  // b*3 + c, 0..191

  // ---- Noise scale from quality (uniform scalar load).
  int q = *qptr;
  float qf = (q < 50) ? 5000.0f / (float)q : fmaf(-2.0f, (float)q, 200.0f);
  float ns = qf * 0.001f;
  float sp = ns * 0.02f;                   // per-pixel sigma
  float sb = ns * 0.08f;                   // boundary sigma: 0.01*sqrt(64)

  // ---- Per-pixel noise: 2 threefry calls -> 4 normals.
  uint32_t p0 = e0 >> 1;
  U2 r0 = threefry2x32(KP0, KP1, p0,      0x03707344u);
  U2 r1 = threefry2x32(KP0, KP1, p0 + 1u, 0x03707344u);
  float pn0 = u32_normal(r0.a), pn1 = u32_normal(r0.b);
  float pn2 = u32_normal(r1.a), pn3 = u32_normal(r1.b);

  // ---- Row-boundary noise (rows 8,16,...,504): shape (B,C,63,W).
  float rn0 = 0.f, rn1 = 0.f, rn2 = 0.f, rn3 = 0.f;
  if (((h & 7u) == 0u) && (h != 0u)) {
    uint32_t rbase = (bc * 63u + ((h >> 3) - 1u)) * Wd + w;   // even
    uint32_t rp = rbase >> 1;
    U2 a = threefry2x32(KR0, KR1, rp,      0x13198a2eu);
    U2 b = threefry2x32(KR0, KR1, rp + 1u, 0x13198a2eu);
    rn0 = u32_normal(a.a); rn1 = u32_normal(a.b);
    rn2 = u32_normal(b.a); rn3 = u32_normal(b.b);
  }

  // ---- Col-boundary noise: only component 0 of the quad can sit on a
  // boundary col (multiples of 8). Shape (B,C,H,63).
  float cn0 = 0.f;
  if (((w & 7u) == 0u) && (w != 0u)) {
    uint32_t ci = (bc * Hd + h) * 63u + ((w >> 3) - 1u);
    U2 a = threefry2x32(KC0, KC1, ci >> 1, 0x243f6a88u);
    cn0 = u32_normal((ci & 1u) ? a.b : a.a);
  }

  // ---- Drain the async copy, read our quad from LDS, combine, clip, store.
  asm volatile("s_wait_asynccnt 0x0" ::: "memory");
  v4f v = tile[threadIdx.x];

  v4f o;
  o.x = clamp01(v.x + fmaf(sp, pn0, fmaf(sb, rn0, sb * cn0)));
  o.y = clamp01(v.y + fmaf(sp, pn1, sb * rn1));
  o.z = clamp01(v.z + fmaf(sp, pn2, sb * rn2));
  o.w = clamp01(v.w + fmaf(sp, pn3, sb * rn3));

  // Output is never re-read: non-temporal 128-bit store, keep L2 for input.
  __builtin_nontemporal_store(o, (v4f*)(out + e0));
}

extern "C" void kernel_launch(void* const* d_in, const int* in_sizes, int n_in,
                              void* d_out, int out_size, void* d_ws, size_t ws_size,
                              hipStream_t stream) {
  const float* img = (const float*)d_in[0];
  const int* quality = (const int*)d_in[1];
  float* out = (float*)d_out;

  uint32_t n = (uint32_t)out_size;         // 64*3*512*512 = 50,331,648
  uint32_t n4 = n >> 2;                    // one thread per float4
  uint32_t blocks = (n4 + TPB - 1) / TPB;  // 49,152 blocks

  JPEGCompression_25881472926199_kernel<<<dim3(blocks), dim3(TPB), 0, stream>>>(
      img, quality, out, n4);
}